// GaussianVoxelizer_23837068493132
// MI455X (gfx1250) — compile-verified
//
#include <hip/hip_runtime.h>

// CDNA5 / gfx1250 implementation of the Gaussian voxelizer (softmax scatter).
//
// Memory-bound: ~450-550 MB of traffic => ~20-25 us floor at 23.3 TB/s HBM.
// Pipeline:
//   1) zero output (328 MB, float4 stores)
//   2) init per-voxel state (encoded -inf max, zero sum)
//   3) per-point: flat voxel index + atomic segment-max (order-preserving u32)
//   4) per-point: exp(conf - vmax) + atomic float segment-sum
//   5) per-point: weight = exp / (sum + eps)
//   6) per 16x16 tile wave: D = diag(w) x F via V_WMMA_F32_16X16X4_F32
//      (bit-exact: each product is exactly w_m * F[m][n] + exact zeros),
//      then 8 global float atomic-adds per lane into out[flat[m]*C + n].

typedef __attribute__((ext_vector_type(2))) float v2f;
typedef __attribute__((ext_vector_type(8))) float v8f;

namespace {
constexpr int   kH = 200, kW = 200, kD = 16;
constexpr int   kS = kH * kW * kD;      // 640000 voxels
constexpr float kEps  = 1e-6f;
constexpr float kVox  = 0.4f;
constexpr float kMinX = -40.f, kMinY = -40.f, kMinZ = -1.f;
}

// Order-preserving float->uint map so unsigned atomicMax == float max.
__device__ __forceinline__ unsigned encOrd(float f) {
  unsigned b = __float_as_uint(f);
  return (b & 0x80000000u) ? ~b : (b | 0x80000000u);
}
__device__ __forceinline__ float decOrd(unsigned e) {
  unsigned b = (e & 0x80000000u) ? (e ^ 0x80000000u) : ~e;
  return __uint_as_float(b);
}

__global__ void k_zero_out(float4* __restrict__ out, long n4) {
  long i = (long)blockIdx.x * blockDim.x + threadIdx.x;
  if (i < n4) out[i] = make_float4(0.f, 0.f, 0.f, 0.f);
}

__global__ void k_init_vox(unsigned* __restrict__ vmax, float* __restrict__ wsum, int s) {
  int i = blockIdx.x * blockDim.x + threadIdx.x;
  if (i < s) { vmax[i] = 0u; wsum[i] = 0.f; }   // enc(-inf)=0x007FFFFF > 0
}

__global__ void k_point_prep(const float* __restrict__ means, const float* __restrict__ opac,
                             int* __restrict__ flat, unsigned* __restrict__ vmax, int n) {
  int i = blockIdx.x * blockDim.x + threadIdx.x;
  if (i >= n) return;
  float x = means[3 * i + 0], y = means[3 * i + 1], z = means[3 * i + 2];
  int vx = (int)rintf((x - kMinX) / kVox);   // rintf == RNE == jnp.round
  int vy = (int)rintf((y - kMinY) / kVox);
  int vz = (int)rintf((z - kMinZ) / kVox);
  vx = min(max(vx, 0), kH - 1);
  vy = min(max(vy, 0), kW - 1);
  vz = min(max(vz, 0), kD - 1);
  int f = vx * (kW * kD) + vy * kD + vz;
  flat[i] = f;
  atomicMax(&vmax[f], encOrd(opac[i]));      // global_atomic_max_u32
}

__global__ void k_exp_sum(const float* __restrict__ opac, const int* __restrict__ flat,
                          const unsigned* __restrict__ vmax, float* __restrict__ cexp,
                          float* __restrict__ wsum, int n) {
  int i = blockIdx.x * blockDim.x + threadIdx.x;
  if (i >= n) return;
  int f = flat[i];
  float e = expf(opac[i] - decOrd(vmax[f]));
  cexp[i] = e;
  __hip_atomic_fetch_add(&wsum[f], e, __ATOMIC_RELAXED, __HIP_MEMORY_SCOPE_AGENT);
}

__global__ void k_weight(const float* __restrict__ cexp, const float* __restrict__ wsum,
                         const int* __restrict__ flat, float* __restrict__ wgt, int n) {
  int i = blockIdx.x * blockDim.x + threadIdx.x;
  if (i < n) wgt[i] = cexp[i] / (wsum[flat[i]] + kEps);
}

// One wave (32 lanes) per 16-point x 16-channel tile.
// A (16x16 diag(w), split into four 16x4 K-chunks): lane holds row m=lane&15,
//   VGPR j of chunk q holds K = 4q + 2*(lane>>4) + j  (ISA 7.12.2, 32-bit A 16x4).
// B (4x16 = feature rows, mirrored layout): lane holds col n=lane&15, same K map.
// D/C (16x16 f32): lane holds col n=lane&15, VGPR r holds row m = r + 8*(lane>>4).
__global__ void __launch_bounds__(256)
k_scatter_wmma(const float* __restrict__ feats, const float* __restrict__ wgt,
               const int* __restrict__ flat, float* __restrict__ out,
               int n, int c, int nPointTiles, int nChanTiles) {
  int wave = (int)((blockIdx.x * (long)blockDim.x + threadIdx.x) >> 5);
  int lane = threadIdx.x & 31;
  long nTiles = (long)nPointTiles * nChanTiles;
  if (wave >= nTiles) return;                // wave-uniform exit; EXEC all-1s below

  int pt = wave / nChanTiles;
  int ct = wave - pt * nChanTiles;
  int pb = pt * 16;                          // point tile base
  int cb = ct * 16;                          // channel tile base
  int mlane = lane & 15;
  int khi   = lane >> 4;                     // 0 or 1

  // Weight for row m = mlane (select, not divergence: EXEC stays full).
  int pm  = pb + mlane;
  int pmc = pm < n ? pm : n - 1;
  float w = (pm < n) ? wgt[pmc] : 0.f;

  // Hoist the 8 flat-index gathers (rows r + 8*khi) so their latency hides
  // under the WMMA chain. Clamped loads are safe; tail rows contribute w=0.
  int flt[8];
#pragma unroll
  for (int r = 0; r < 8; ++r) {
    int p = pb + r + 8 * khi;
    flt[r] = flat[p < n ? p : n - 1];
  }

  v8f acc = {};
#pragma unroll
  for (int q = 0; q < 4; ++q) {
    int k0 = 4 * q + 2 * khi;
    v2f a, b;
    a.x = (k0     == mlane) ? w : 0.f;
    a.y = (k0 + 1 == mlane) ? w : 0.f;
    int r0 = pb + k0;
    int r1 = r0 + 1;
    r0 = r0 < n ? r0 : n - 1;                // clamped: zero weight kills tail rows
    r1 = r1 < n ? r1 : n - 1;
    b.x = feats[(long)r0 * c + cb + mlane];
    b.y = feats[(long)r1 * c + cb + mlane];
    // D = A x B + C  (exact: diag products only, plus exact zeros)
    acc = __builtin_amdgcn_wmma_f32_16x16x4_f32(
        /*neg_a=*/false, a, /*neg_b=*/false, b,
        /*c_mod=*/(short)0, acc, /*reuse_a=*/false, /*reuse_b=*/false);
  }

  // Scatter-add D rows. Fast path (always taken for N%16==0): unconditional
  // atomics -> compiler can emit a clean burst of 8 global_atomic_add_f32.
  long colOff = cb + mlane;
  if (pb + 16 <= n) {
#pragma unroll
    for (int r = 0; r < 8; ++r) {
      __hip_atomic_fetch_add(&out[(long)flt[r] * c + colOff], acc[r],
                             __ATOMIC_RELAXED, __HIP_MEMORY_SCOPE_AGENT);
    }
  } else {
#pragma unroll
    for (int r = 0; r < 8; ++r) {
      if (pb + r + 8 * khi < n) {
        __hip_atomic_fetch_add(&out[(long)flt[r] * c + colOff], acc[r],
                               __ATOMIC_RELAXED, __HIP_MEMORY_SCOPE_AGENT);
      }
    }
  }
}

extern "C" void kernel_launch(void* const* d_in, const int* in_sizes, int n_in,
                              void* d_out, int out_size, void* d_ws, size_t ws_size,
                              hipStream_t stream) {
  const float* means = (const float*)d_in[0];   // (N,3)
  const float* opac  = (const float*)d_in[1];   // (N,1)
  const float* feats = (const float*)d_in[2];   // (N,1,C)
  float* out = (float*)d_out;                   // (H,W,D,C)

  const int n = in_sizes[0] / 3;
  const int c = (n > 0) ? (in_sizes[2] / n) : 0;

  // Workspace layout (~7.5 MB total)
  char* ws = (char*)d_ws;
  size_t off = 0;
  int*      flat = (int*)(ws + off);      off += (size_t)n * 4;
  off = (off + 15) & ~(size_t)15;
  unsigned* vmax = (unsigned*)(ws + off); off += (size_t)kS * 4;
  float*    wsum = (float*)(ws + off);    off += (size_t)kS * 4;
  float*    cexp = (float*)(ws + off);    off += (size_t)n * 4;
  float*    wgt  = (float*)(ws + off);    off += (size_t)n * 4;

  long n4 = (long)out_size / 4;           // out_size = S*C, divisible by 4
  k_zero_out<<<(int)((n4 + 255) / 256), 256, 0, stream>>>((float4*)out, n4);
  k_init_vox<<<(kS + 255) / 256, 256, 0, stream>>>(vmax, wsum, kS);

  int pblk = (n + 255) / 256;
  k_point_prep<<<pblk, 256, 0, stream>>>(means, opac, flat, vmax, n);
  k_exp_sum<<<pblk, 256, 0, stream>>>(opac, flat, vmax, cexp, wsum, n);
  k_weight<<<pblk, 256, 0, stream>>>(cexp, wsum, flat, wgt, n);

  int ptiles = (n + 15) / 16;
  int ctiles = c / 16;                    // C=128 -> 8
  long waves   = (long)ptiles * ctiles;   // 100,000 waves
  long threads = waves * 32;
  k_scatter_wmma<<<(int)((threads + 255) / 256), 256, 0, stream>>>(
      feats, wgt, flat, out, n, c, ptiles, ctiles);
}